// DigitCapsules_14800457302030
// MI455X (gfx1250) — compile-verified
//
#include <hip/hip_runtime.h>
#include <hip/hip_bf16.h>

typedef __attribute__((ext_vector_type(2))) float v2f;
typedef __attribute__((ext_vector_type(8))) float v8f;
typedef __attribute__((ext_vector_type(4))) unsigned int v4u;
typedef __attribute__((ext_vector_type(8))) int v8i;
typedef __attribute__((ext_vector_type(4))) int v4i;

#define BATCH 64
#define IC    2048
#define NC    32
#define OD    16
#define IDIM  8
#define JODIM 512                       // NC*OD
#define UB_STRIDE ((size_t)IC * JODIM)  // u_hat stride per batch element (1,048,576)

// ---------------------------------------------------------------------------
// TDM: issue a 2D tensor_load_to_lds (data_size = 4B elements).
// D# packed per CDNA5 ISA 8.3/8.4: group0 = {flags, lds_addr, global_addr, type=2},
// group1 = {data_size, tensor_dim0/1, tile_dim0/1, tensor_dim0_stride}.
// tile_dim2 = 0 -> groups 2/3 unused (zero-filled).
// ---------------------------------------------------------------------------
__device__ static inline void tdm_load_2d(unsigned lds_off, const void* gaddr,
                                          unsigned tensor_d0, unsigned tensor_d1,
                                          unsigned tile_d0, unsigned tile_d1,
                                          unsigned long long d0_stride_elems) {
  const unsigned long long ga = (unsigned long long)(uintptr_t)gaddr;
  v4u g0;
  g0[0] = 1u;                                                  // count=1, user D#
  g0[1] = lds_off;                                             // LDS byte address
  g0[2] = (unsigned)(ga & 0xFFFFFFFFu);                        // global_addr[31:0]
  g0[3] = (unsigned)((ga >> 32) & 0x1FFFFFFu) | 0x80000000u;   // addr[56:32] | type=2
  v8i g1;
  g1[0] = 0x00020000;                                          // data_size = 2 (4B)
  g1[1] = (int)((tensor_d0 & 0xFFFFu) << 16);                  // tensor_dim0[15:0]
  g1[2] = (int)(((tensor_d1 & 0xFFFFu) << 16) | ((tensor_d0 >> 16) & 0xFFFFu));
  g1[3] = (int)(((tile_d0 & 0xFFFFu) << 16) | ((tensor_d1 >> 16) & 0xFFFFu));
  g1[4] = (int)(tile_d1 & 0xFFFFu);                            // tile_dim1 (tile_dim2=0)
  g1[5] = (int)(d0_stride_elems & 0xFFFFFFFFull);              // tensor_dim0_stride lo
  g1[6] = (int)((d0_stride_elems >> 32) & 0xFFFFull);          // stride hi (dim1_stride=0)
  g1[7] = 0;
  const v4i gz = {0, 0, 0, 0};
#if __clang_major__ >= 23
  const v8i gz8 = {0, 0, 0, 0, 0, 0, 0, 0};
  __builtin_amdgcn_tensor_load_to_lds(g0, g1, gz, gz, gz8, 0);
#else
  __builtin_amdgcn_tensor_load_to_lds(g0, g1, gz, gz, 0);
#endif
}

// ---------------------------------------------------------------------------
// Kernel 1: u_hat[b,i,jo] = sum_d W[i,jo,d] * x[b,i,d]   via V_WMMA_F32_16X16X4_F32
// One block per i. A = x tile (M=batch, K=d), B = W_i^T (K=d, N=jo).
// W_i (16KB, 1D) and x[:,i,:] (64 rows x 32B, stride 64KB) staged by TDM.
// ---------------------------------------------------------------------------
__global__ __launch_bounds__(256) void capsule_uhat_wmma(
    const float* __restrict__ x, const float* __restrict__ W, float* __restrict__ u) {
  __shared__ float Wl[JODIM * IDIM];  // 16 KB: W_i rows [jo][d]
  __shared__ float Xl[BATCH * IDIM];  // 2 KB : x[:, i, :] rows [b][d]
  const int i = blockIdx.x;
  const int t = threadIdx.x;
  const int wave = t >> 5, lane = t & 31;

  if (wave == 0) {
    // W_i: contiguous 4096 floats as one 1D tile.
    tdm_load_2d((unsigned)(uintptr_t)(void*)Wl, W + (size_t)i * (JODIM * IDIM),
                /*tensor_d0=*/4096, /*tensor_d1=*/1, /*tile_d0=*/4096, /*tile_d1=*/1,
                /*d0_stride=*/4096ull);
    // x[:,i,:]: 64 rows of 8 floats, row stride IC*IDIM elements -> packs to Xl[b][d].
    tdm_load_2d((unsigned)(uintptr_t)(void*)Xl, x + (size_t)i * IDIM,
                /*tensor_d0=*/IDIM, /*tensor_d1=*/BATCH, /*tile_d0=*/IDIM, /*tile_d1=*/BATCH,
                /*d0_stride=*/(unsigned long long)IC * IDIM);
    __builtin_amdgcn_s_wait_tensorcnt(0);
  }
  __syncthreads();

  const int ln = lane & 15;
  const int kh = (lane < 16) ? 0 : 2;  // A/B 16x4 f32 layout: lanes 16-31 hold K=2,3

#pragma unroll 1
  for (int it = 0; it < 16; ++it) {   // 8 waves x 16 tiles = 128 tiles (4 Mt x 32 Nt)
    const int tile = wave + 8 * it;
    const int Mt = tile & 3;          // batch tile (wave-invariant -> A frag hoisted)
    const int Nt = tile >> 2;         // jo tile
    const int bl = Mt * 16 + ln;      // batch row for A frag
    const int jo = Nt * 16 + ln;      // jo col for B frag
    const v2f a0 = *(const v2f*)(Xl + bl * IDIM + kh);
    const v2f a1 = *(const v2f*)(Xl + bl * IDIM + kh + 4);
    const v2f b0 = *(const v2f*)(Wl + jo * IDIM + kh);
    const v2f b1 = *(const v2f*)(Wl + jo * IDIM + kh + 4);
    v8f c = {};
    c = __builtin_amdgcn_wmma_f32_16x16x4_f32(false, a0, false, b0, (short)0, c, false, false);
    c = __builtin_amdgcn_wmma_f32_16x16x4_f32(false, a1, false, b1, (short)0, c, false, false);
    // C layout: VGPR v holds M = v (lanes 0-15) / v+8 (lanes 16-31), N = lane&15.
    const int mbase = Mt * 16 + ((lane < 16) ? 0 : 8);
    float* ubase = u + (size_t)i * JODIM + jo;
#pragma unroll
    for (int v = 0; v < 8; ++v) {
      ubase[(size_t)(mbase + v) * UB_STRIDE] = c[v];  // half-wave writes 64B-contiguous jo
    }
  }
}

// ---------------------------------------------------------------------------
// Kernel 2: fused routing pass. One pass over u_hat per iteration.
// MODE 0: b=0 -> c=1/32, accumulate s-partials only.
// MODE 1: b_new = u.v (b_old=0), write b, softmax, accumulate.
// MODE 2: b_new = b_old + u.v (read b, no write), softmax, accumulate.
// Grid: (16 i-chunks, 64 batch). Lane l owns capsule j=l.
// ---------------------------------------------------------------------------
template <int MODE>
__global__ __launch_bounds__(256) void routing_pass(
    const float* __restrict__ u, const float* __restrict__ vprev,
    float* __restrict__ bij, float* __restrict__ partial) {
  __shared__ float vjT[JODIM];      // v transposed [o][j] -> conflict-free reads
  __shared__ float sacc[8][JODIM];  // per-wave partial s, o-major [o*32+j]
  const int b = blockIdx.y, chunk = blockIdx.x;
  const int t = threadIdx.x, wave = t >> 5, lane = t & 31;

  if constexpr (MODE >= 1) {
    const int g0 = t, g1 = t + 256;
    const float a0 = vprev[b * JODIM + g0];
    const float a1 = vprev[b * JODIM + g1];
    vjT[(g0 & 15) * 32 + (g0 >> 4)] = a0;
    vjT[(g1 & 15) * 32 + (g1 >> 4)] = a1;
  }
#pragma unroll
  for (int o = 0; o < 16; ++o) sacc[wave][o * 32 + lane] = 0.f;
  __syncthreads();

  const float* ub = u + (size_t)b * UB_STRIDE;
#pragma unroll 1
  for (int it = 0; it < 16; ++it) {
    const int i = chunk * 128 + wave + 8 * it;
    float uo[16];
    const float4* ur4 = (const float4*)(ub + (size_t)i * JODIM + lane * 16);
    *(float4*)(uo + 0)  = ur4[0];   // wave covers one full 2KB u row -> coalesced b128
    *(float4*)(uo + 4)  = ur4[1];
    *(float4*)(uo + 8)  = ur4[2];
    *(float4*)(uo + 12) = ur4[3];
    if (it < 15)  // prefetch next row for this wave (8 i's ahead)
      __builtin_prefetch((const void*)(ub + (size_t)(i + 8) * JODIM + lane * 16), 0, 1);

    float c;
    if constexpr (MODE == 0) {
      c = 1.0f / 32.0f;
    } else {
      float dv = 0.f;
#pragma unroll
      for (int o = 0; o < 16; ++o) dv = fmaf(uo[o], vjT[o * 32 + lane], dv);
      float bold = 0.f;
      const int bidx = (b * IC + i) * NC + lane;
      if constexpr (MODE == 2) bold = bij[bidx];
      const float bnew = bold + dv;
      if constexpr (MODE == 1) bij[bidx] = bnew;
      // softmax over the 32 capsules (one per lane)
      float mx = bnew;
#pragma unroll
      for (int off = 16; off > 0; off >>= 1) mx = fmaxf(mx, __shfl_xor(mx, off, 32));
      const float e = __expf(bnew - mx);
      float sum = e;
#pragma unroll
      for (int off = 16; off > 0; off >>= 1) sum += __shfl_xor(sum, off, 32);
      c = e / sum;
    }
#pragma unroll
    for (int o = 0; o < 16; ++o) sacc[wave][o * 32 + lane] += c * uo[o];
  }
  __syncthreads();

  // deterministic cross-wave reduction -> partial[b][chunk][o*32+j]
  float r0 = 0.f, r1 = 0.f;
#pragma unroll
  for (int w = 0; w < 8; ++w) { r0 += sacc[w][t]; r1 += sacc[w][t + 256]; }
  float* p = partial + ((size_t)b * 16 + chunk) * JODIM;
  p[t]       = r0;
  p[t + 256] = r1;
}

// ---------------------------------------------------------------------------
// Kernel 3: reduce 16 chunk-partials, squash, emit v[b,j,o]. One block per b.
// ---------------------------------------------------------------------------
__global__ __launch_bounds__(512) void reduce_squash(
    const float* __restrict__ partial, float* __restrict__ vout) {
  const int b = blockIdx.x, t = threadIdx.x;  // t = j*16 + o
  const int j = t >> 4, o = t & 15;
  const float* p = partial + (size_t)b * 16 * JODIM + o * 32 + j;
  float s = 0.f;
#pragma unroll
  for (int ch = 0; ch < 16; ++ch) s += p[ch * JODIM];
  // sum of squares over o: 16-lane groups (lane = o + 16*(j&1))
  float s2 = s * s;
#pragma unroll
  for (int off = 8; off > 0; off >>= 1) s2 += __shfl_xor(s2, off, 32);
  const float scale = (s2 / (1.f + s2)) * rsqrtf(s2 + 1e-8f);
  vout[b * JODIM + t] = s * scale;
}

// ---------------------------------------------------------------------------
extern "C" void kernel_launch(void* const* d_in, const int* in_sizes, int n_in,
                              void* d_out, int out_size, void* d_ws, size_t ws_size,
                              hipStream_t stream) {
  (void)in_sizes; (void)n_in; (void)out_size; (void)ws_size;
  const float* x = (const float*)d_in[0];  // [64, 2048, 8]
  const float* W = (const float*)d_in[1];  // [1, 2048, 32, 16, 8]
  float* out = (float*)d_out;              // [64, 32, 16]

  float* u       = (float*)d_ws;                          // 64*2048*512  f32 (256 MB)
  float* bij     = u + (size_t)BATCH * IC * JODIM;        // 64*2048*32   f32 (16 MB)
  float* partial = bij + (size_t)BATCH * IC * NC;         // 64*16*512    f32 (2 MB)
  float* vbuf    = partial + (size_t)BATCH * 16 * JODIM;  // 64*512       f32

  capsule_uhat_wmma<<<dim3(IC), dim3(256), 0, stream>>>(x, W, u);

  const dim3 rgrid(16, BATCH);
  routing_pass<0><<<rgrid, 256, 0, stream>>>(u, nullptr, bij, partial);  // iter 1: c uniform
  reduce_squash<<<BATCH, 512, 0, stream>>>(partial, vbuf);               // -> v1
  routing_pass<1><<<rgrid, 256, 0, stream>>>(u, vbuf, bij, partial);     // b1 = u.v1, c2, s2
  reduce_squash<<<BATCH, 512, 0, stream>>>(partial, vbuf);               // -> v2
  routing_pass<2><<<rgrid, 256, 0, stream>>>(u, vbuf, bij, partial);     // b2 = b1+u.v2, c3, s3
  reduce_squash<<<BATCH, 512, 0, stream>>>(partial, out);                // -> v3 (output)
}